// Detector_54116587929726
// MI455X (gfx1250) — compile-verified
//
#include <hip/hip_runtime.h>
#include <math.h>

typedef float v2f __attribute__((ext_vector_type(2)));
typedef float v8f __attribute__((ext_vector_type(8)));

#define BATCH     4
#define NPROP     2000
#define PRE       1000
#define MAXOUT    128
#define MIN_VOX   (28*28*28)   /* 21952 */
#define MOUT_VOX  (32*32*32)   /* 32768 */
#define NMS_THR   0.3f
#define SORT_N    2048

// Monotonic unsigned mapping of float ordering (handles -inf sentinel).
__device__ __forceinline__ unsigned fkey(float f) {
    unsigned u = __float_as_uint(f);
    return (u & 0x80000000u) ? ~u : (u | 0x80000000u);
}

// ---------------------------------------------------------------------------
// Kernel 1: per-batch top-K sort + box regression + greedy 3D NMS.
// grid = BATCH blocks, 1024 threads (32 wave32 waves). Everything in LDS.
// Argmax per NMS round: packed (scorekey << 32 | ~idx) max-reduction done
// with 5 wave32 shuffle steps + one cross-wave LDS stage (3 barriers/round).
// ---------------------------------------------------------------------------
__global__ __launch_bounds__(1024)
void sort_nms_kernel(const float* __restrict__ proposals,
                     const float* __restrict__ scores,
                     const float* __restrict__ deltas,
                     float* __restrict__ out,         // full output buffer
                     int* __restrict__ ws_src)        // [BATCH*MAXOUT]
{
    __shared__ float s_score[SORT_N];
    __shared__ int   s_idx[SORT_N];
    __shared__ float s_box[PRE * 6];
    __shared__ int   s_sup[PRE];
    __shared__ unsigned long long s_red[32];
    __shared__ int   s_kidx[MAXOUT];
    __shared__ int   s_kval[MAXOUT];
    __shared__ int   s_sel;

    const int b = blockIdx.x;
    const int t = threadIdx.x;
    const int wave = t >> 5;
    const int lane = t & 31;
    const float NEG = -__builtin_inff();

    // ---- load scores (pad to 2048 with -inf) ----
    for (int e = t; e < SORT_N; e += 1024) {
        s_score[e] = (e < NPROP) ? scores[b * NPROP + e] : NEG;
        s_idx[e]   = e;
    }
    __syncthreads();

    // ---- bitonic sort, descending, stable (index ascending on ties) ----
    for (int k = 2; k <= SORT_N; k <<= 1) {
        for (int j = k >> 1; j > 0; j >>= 1) {
            for (int e = t; e < SORT_N; e += 1024) {
                int p = e ^ j;
                if (p > e) {
                    float se = s_score[e], sp = s_score[p];
                    int   ie = s_idx[e],   ip = s_idx[p];
                    bool e_after = (se < sp) || (se == sp && ie > ip);
                    bool desc    = ((e & k) == 0);
                    if (desc ? e_after : !e_after) {
                        s_score[e] = sp; s_score[p] = se;
                        s_idx[e]   = ip; s_idx[p]   = ie;
                    }
                }
            }
            __syncthreads();
        }
    }

    // ---- box regression on top PRE proposals ----
    if (t < PRE) {
        int oi = s_idx[t];
        const float* pb = proposals + (size_t)(b * NPROP + oi) * 6;
        const float* dl = deltas    + (size_t)(b * NPROP + oi) * 6;
        float h = pb[3] - pb[0], w = pb[4] - pb[1], d = pb[5] - pb[2];
        float cy = pb[0] + 0.5f * h + dl[0] * h;
        float cx = pb[1] + 0.5f * w + dl[1] * w;
        float cz = pb[2] + 0.5f * d + dl[2] * d;
        h *= expf(dl[3]); w *= expf(dl[4]); d *= expf(dl[5]);
        float* bx = s_box + t * 6;
        bx[0] = cy - 0.5f * h; bx[1] = cx - 0.5f * w; bx[2] = cz - 0.5f * d;
        bx[3] = cy + 0.5f * h; bx[4] = cx + 0.5f * w; bx[5] = cz + 0.5f * d;
        s_sup[t] = 0;
    }
    __syncthreads();

    // ---- greedy NMS: MAXOUT rounds of argmax + suppression ----
    for (int iter = 0; iter < MAXOUT; ++iter) {
        float v = NEG;
        if (t < PRE && !s_sup[t]) v = s_score[t];
        // pack: high 32 = monotonic score key, low 32 = ~index (max => min idx)
        unsigned long long p =
            ((unsigned long long)fkey(v) << 32) | (unsigned)(~t);
        #pragma unroll
        for (int off = 16; off > 0; off >>= 1) {
            unsigned long long q = __shfl_xor(p, off, 32);
            if (q > p) p = q;
        }
        if (lane == 0) s_red[wave] = p;
        __syncthreads();
        if (wave == 0) {
            unsigned long long q = s_red[lane];
            #pragma unroll
            for (int off = 16; off > 0; off >>= 1) {
                unsigned long long r = __shfl_xor(q, off, 32);
                if (r > q) q = r;
            }
            if (lane == 0) {
                int sel = (int)(~(unsigned)(q & 0xFFFFFFFFull));
                s_sel = sel;
                s_kidx[iter] = sel;
                s_kval[iter] = s_sup[sel] ? 0 : 1;   // valid before suppression
            }
        }
        __syncthreads();
        int sel = s_sel;
        if (t < PRE) {
            const float* a = s_box + sel * 6;
            const float* c = s_box + t * 6;
            float iy = fmaxf(fminf(a[3], c[3]) - fmaxf(a[0], c[0]), 0.f);
            float ix = fmaxf(fminf(a[4], c[4]) - fmaxf(a[1], c[1]), 0.f);
            float iz = fmaxf(fminf(a[5], c[5]) - fmaxf(a[2], c[2]), 0.f);
            float inter = iy * ix * iz;
            float v1 = (a[3] - a[0]) * (a[4] - a[1]) * (a[5] - a[2]);
            float v2 = (c[3] - c[0]) * (c[4] - c[1]) * (c[5] - c[2]);
            float iou = inter / (v1 + v2 - inter + 1e-8f);
            if (iou > NMS_THR) s_sup[t] = 1;
            if (t == sel)      s_sup[t] = 1;
        }
        __syncthreads();
    }

    // ---- emit boxes, batch indices, and mask gather table ----
    const int idx_base = BATCH * MAXOUT * 6 + BATCH * MAXOUT * MOUT_VOX;
    for (int k = t; k < MAXOUT; k += 1024) {
        int ki = s_kidx[k];
        int kv = s_kval[k];
        float vf = kv ? 1.f : 0.f;
        float* ob = out + (size_t)(b * MAXOUT + k) * 6;
        #pragma unroll
        for (int j = 0; j < 6; ++j) ob[j] = s_box[ki * 6 + j] * vf;
        out[idx_base + b * MAXOUT + k] = (float)b;
        ws_src[b * MAXOUT + k] = kv ? (b * NPROP + s_idx[ki]) : -1;
    }
}

// ---------------------------------------------------------------------------
// Trilinear (half-pixel, edge-clamped) 28 -> 32 interpolation weight.
// ---------------------------------------------------------------------------
__device__ __forceinline__ float resize_w(int n, int k) {
    float pos = (n + 0.5f) * (28.0f / 32.0f) - 0.5f;
    float fl  = floorf(pos);
    int   i0  = (int)fl;
    float f   = pos - fl;
    int ia = i0 < 0 ? 0 : (i0 > 27 ? 27 : i0);
    int i1 = i0 + 1;
    int ib = i1 < 0 ? 0 : (i1 > 27 ? 27 : i1);
    float w = 0.f;
    if (k == ia) w += 1.f - f;
    if (k == ib) w += f;
    return w;
}

// ---------------------------------------------------------------------------
// Kernel 2: per-mask 28^3 -> 32^3 trilinear resize as 3 separable GEMMs
// on V_WMMA_F32_16X16X4_F32 (full f32 precision, K=28 = 7 x K4 steps).
// Mask ingress uses gfx1250 async global->LDS (ASYNCcnt) to skip the VGPR
// round-trip; it overlaps with on-the-fly weight-matrix generation.
// LDS: Wm 896f | bufA 28672f (input, then pass2 out) | bufB 25088f.
// ---------------------------------------------------------------------------
__global__ __launch_bounds__(256)
void mask_resize_kernel(const float* __restrict__ masks,
                        const int* __restrict__ src_idx,
                        float* __restrict__ out_masks)
{
    extern __shared__ float smem[];
    float* Wm   = smem;                  // 896
    float* bufA = smem + 896;            // 28672
    float* bufB = smem + 896 + 28672;    // 25088

    const int m = blockIdx.x;
    const int t = threadIdx.x;
    float* omask = out_masks + (size_t)m * MOUT_VOX;

    const int src = src_idx[m];
    if (src < 0) {                        // invalid slot -> zero mask
        for (int i = t; i < MOUT_VOX; i += 256) omask[i] = 0.f;
        return;
    }

    // ---- async DMA: 21952 floats = 5488 x B128 straight into LDS ----
    {
        const float* im = masks + (size_t)src * MIN_VOX;
        unsigned lbase = (unsigned)(unsigned long long)(void*)bufA; // LDS byte offset
        for (int i = t; i < MIN_VOX / 4; i += 256) {
            unsigned long long ga = (unsigned long long)(im + i * 4);
            unsigned la = lbase + (unsigned)i * 16u;
            asm volatile("global_load_async_to_lds_b128 %0, %1, off"
                         :: "v"(la), "v"(ga) : "memory");
        }
    }
    // weight matrix while the DMA flies
    for (int i = t; i < 32 * 28; i += 256) Wm[i] = resize_w(i / 28, i % 28);
    asm volatile("s_wait_asynccnt 0x0" ::: "memory");
    __syncthreads();

    const int wave = t >> 5;              // wave32
    const int lane = t & 31;
    const int lo   = lane & 15;           // M (A) / N (B,C) lane slot
    const int hi   = lane >> 4;           // K-pair select

    // ---- pass 1: resize x.  C[784,32] = In[784,28] x W^T[28,32] -> bufB
    for (int tile = wave; tile < 98; tile += 8) {
        int tm = tile >> 1, tn = tile & 1;
        const float* ar = bufA + (tm * 16 + lo) * 28;      // A row (M = lo)
        const float* wr = Wm   + (tn * 16 + lo) * 28;      // B col = W row n
        v8f c = {};
        #pragma unroll
        for (int kc = 0; kc < 7; ++kc) {
            int ka = kc * 4 + hi * 2;
            v2f a; a.x = ar[ka]; a.y = ar[ka + 1];
            v2f bb; bb.x = wr[ka]; bb.y = wr[ka + 1];
            c = __builtin_amdgcn_wmma_f32_16x16x4_f32(false, a, false, bb,
                                                      (short)0, c, false, false);
        }
        #pragma unroll
        for (int r = 0; r < 8; ++r) {
            int gm = tm * 16 + r + hi * 8;                 // C row
            bufB[gm * 32 + tn * 16 + lo] = c[r];
        }
    }
    __syncthreads();

    // ---- pass 2: resize y.  per z: C[32,32] = W[32,28] x B1z[28,32] -> bufA
    for (int tile = wave; tile < 112; tile += 8) {
        int z = tile >> 2, tm = (tile >> 1) & 1, tn = tile & 1;
        const float* wr = Wm + (tm * 16 + lo) * 28;        // A from W
        const float* bz = bufB + z * 28 * 32 + tn * 16 + lo;
        v8f c = {};
        #pragma unroll
        for (int kc = 0; kc < 7; ++kc) {
            int ka = kc * 4 + hi * 2;
            v2f a; a.x = wr[ka]; a.y = wr[ka + 1];
            v2f bb; bb.x = bz[ka * 32]; bb.y = bz[(ka + 1) * 32];
            c = __builtin_amdgcn_wmma_f32_16x16x4_f32(false, a, false, bb,
                                                      (short)0, c, false, false);
        }
        #pragma unroll
        for (int r = 0; r < 8; ++r) {
            int yq = tm * 16 + r + hi * 8;
            bufA[(z * 32 + yq) * 32 + tn * 16 + lo] = c[r];
        }
    }
    __syncthreads();

    // ---- pass 3: resize z.  C[32,1024] = W[32,28] x B2[28,1024] -> global
    for (int tile = wave; tile < 128; tile += 8) {
        int tm = tile >> 6, tn = tile & 63;
        const float* wr = Wm + (tm * 16 + lo) * 28;
        const float* bz = bufA + tn * 16 + lo;
        v8f c = {};
        #pragma unroll
        for (int kc = 0; kc < 7; ++kc) {
            int ka = kc * 4 + hi * 2;
            v2f a; a.x = wr[ka]; a.y = wr[ka + 1];
            v2f bb; bb.x = bz[ka * 1024]; bb.y = bz[(ka + 1) * 1024];
            c = __builtin_amdgcn_wmma_f32_16x16x4_f32(false, a, false, bb,
                                                      (short)0, c, false, false);
        }
        #pragma unroll
        for (int r = 0; r < 8; ++r) {
            int zq = tm * 16 + r + hi * 8;
            omask[zq * 1024 + tn * 16 + lo] = c[r];
        }
    }
}

extern "C" void kernel_launch(void* const* d_in, const int* in_sizes, int n_in,
                              void* d_out, int out_size, void* d_ws, size_t ws_size,
                              hipStream_t stream) {
    const float* proposals = (const float*)d_in[0];
    const float* scores    = (const float*)d_in[1];
    const float* deltas    = (const float*)d_in[2];
    const float* masks     = (const float*)d_in[3];
    float* out   = (float*)d_out;
    int*   wsrc  = (int*)d_ws;

    sort_nms_kernel<<<BATCH, 1024, 0, stream>>>(proposals, scores, deltas, out, wsrc);

    const size_t smem = (size_t)(896 + 28672 + 25088) * sizeof(float); // 218624 B
    (void)hipFuncSetAttribute((const void*)mask_resize_kernel,
                              hipFuncAttributeMaxDynamicSharedMemorySize, (int)smem);
    mask_resize_kernel<<<BATCH * MAXOUT, 256, smem, stream>>>(
        masks, wsrc, out + BATCH * MAXOUT * 6);
}